// OffsetPredictor_43654047597203
// MI455X (gfx1250) — compile-verified
//
#include <hip/hip_runtime.h>
#include <math.h>

#define B_DIM 8
#define L_DIM 2048
#define C_DIM 256
#define P_CNT 511   // (2048-8)/4 + 1

typedef __attribute__((ext_vector_type(2))) float v2f;
typedef __attribute__((ext_vector_type(8))) float v8f;

// Exact GeLU: x * 0.5 * erfc(-x/sqrt(2)), branch-free.
// erfc via Numerical-Recipes rational-Chebyshev: fractional error < 1.2e-7,
// i.e. fp32-faithful vs libm erff, but with zero EXEC-mask divergence.
__device__ __forceinline__ float gelu_exact(float x) {
    const float a = __builtin_fabsf(x) * 0.70710678118654752440f;     // |x|/sqrt2
    const float t = __builtin_amdgcn_rcpf(__builtin_fmaf(0.5f, a, 1.0f));
    float p =                     0.17087277f;
    p = __builtin_fmaf(t, p, -0.82215223f);
    p = __builtin_fmaf(t, p,  1.48851587f);
    p = __builtin_fmaf(t, p, -1.13520398f);
    p = __builtin_fmaf(t, p,  0.27886807f);
    p = __builtin_fmaf(t, p, -0.18628806f);
    p = __builtin_fmaf(t, p,  0.09678418f);
    p = __builtin_fmaf(t, p,  0.37409196f);
    p = __builtin_fmaf(t, p,  1.00002368f);
    float arg = __builtin_fmaf(t, p, -1.26551223f);
    arg = __builtin_fmaf(-a, a, arg);                                  // -a^2 + ...
    const float ans =
        t * __builtin_amdgcn_exp2f(arg * 1.4426950408889634f);        // v_exp_f32
    const float ec  = (x > 0.0f) ? (2.0f - ans) : ans;                 // erfc(-x/sqrt2)
    return 0.5f * x * ec;
}

__global__ __launch_bounds__(256)
void offset_predictor_kernel(const float* __restrict__ X,
                             const float* __restrict__ W1,
                             const float* __restrict__ b1,
                             const float* __restrict__ W2,
                             const float* __restrict__ b2,
                             float* __restrict__ out)
{
    const int lane  = threadIdx.x & 31;
    const int wave  = threadIdx.x >> 5;
    const int tile  = blockIdx.x * 8 + wave;       // 65,408 tiles total

    const int lid16 = lane & 15;
    const int half  = lane >> 4;

    // tile -> (b, p, c-chunk)
    const int per_b = P_CNT * (C_DIM / 16);        // 8176
    const int b  = tile / per_b;
    const int r0 = tile - b * per_b;
    const int p  = r0 >> 4;
    const int c0 = (r0 & 15) << 4;

    // ---- A fragments: A[m=c, k=s], fp32 16x4 layout.
    // Lane L holds M = L%16, K = 2*(L/16) + v. k-step0: s=0..3, k-step1: s=4..7.
    const float* xb = X + ((size_t)b * L_DIM + (size_t)p * 4) * C_DIM + (c0 + lid16);
    v2f a0, a1;
    {
        const int s0 = 2 * half;
        a0.x = xb[(size_t)(s0    ) * C_DIM];
        a0.y = xb[(size_t)(s0 + 1) * C_DIM];
        a1.x = xb[(size_t)(s0 + 4) * C_DIM];
        a1.y = xb[(size_t)(s0 + 5) * C_DIM];
    }

    // ---- B fragments: B[k, n] = W1[n=j, s], N across lanes 0-15, K split
    // across half-waves. W1 row-major (64,8); s even -> 8B-aligned pairs.
    v2f bw[4][2];
    #pragma unroll
    for (int t = 0; t < 4; ++t) {
        const int j = 16 * t + lid16;
        const float* w1r = W1 + j * 8 + 2 * half;
        bw[t][0].x = w1r[0]; bw[t][0].y = w1r[1];   // s = 2*half + {0,1}
        bw[t][1].x = w1r[4]; bw[t][1].y = w1r[5];   // s = 4 + 2*half + {0,1}
    }

    // ---- accumulators pre-loaded with bias (bias depends only on N = lane)
    v8f acc[4];
    #pragma unroll
    for (int t = 0; t < 4; ++t) {
        const float bias = b1[16 * t + lid16];
        #pragma unroll
        for (int v = 0; v < 8; ++v) acc[t][v] = bias;
    }

    // ---- layer 1: 8 x V_WMMA_F32_16X16X4_F32 (full fp32 precision)
    #pragma unroll
    for (int t = 0; t < 4; ++t) {
        acc[t] = __builtin_amdgcn_wmma_f32_16x16x4_f32(
            false, a0, false, bw[t][0], (short)0, acc[t], false, false);
        acc[t] = __builtin_amdgcn_wmma_f32_16x16x4_f32(
            false, a1, false, bw[t][1], (short)0, acc[t], false, false);
    }

    // ---- exact GeLU on the 32 h-values this lane holds (branch-free)
    #pragma unroll
    for (int t = 0; t < 4; ++t) {
        #pragma unroll
        for (int v = 0; v < 8; ++v) acc[t][v] = gelu_exact(acc[t][v]);
    }

    // ---- layer 2 partials: flat accumulator index i = 2*v + o (per lane)
    float r[16];
    #pragma unroll
    for (int i = 0; i < 16; ++i) r[i] = 0.0f;
    #pragma unroll
    for (int t = 0; t < 4; ++t) {
        const float wa = W2[     16 * t + lid16];   // W2[0, j], j = 16t+lid16
        const float wb = W2[64 + 16 * t + lid16];   // W2[1, j]
        #pragma unroll
        for (int v = 0; v < 8; ++v) {
            const float h = acc[t][v];
            r[2 * v    ] = fmaf(h, wa, r[2 * v    ]);
            r[2 * v + 1] = fmaf(h, wb, r[2 * v + 1]);
        }
    }

    // ---- reduce-scatter butterfly across each 16-lane half: 4 steps,
    // each lane keeps the accumulator half matching its own lane bit and
    // ships the other half. Ends with lane l holding the full sum for i = l.
    #pragma unroll
    for (int d = 8; d >= 1; d >>= 1) {
        const bool hi = (lid16 & d) != 0;
        #pragma unroll
        for (int k = 0; k < d; ++k) {
            const float own  = hi ? r[d + k] : r[k];
            const float send = hi ? r[k]     : r[d + k];
            r[k] = own + __shfl_xor(send, d, 16);
        }
    }

    // i = 2v+o = lid16  ->  output element (m = v + 8*half, o) sits at flat
    // offset 16*half + lid16 = lane: one fully-coalesced store per lane.
    const float res = r[0] + b2[lane & 1];
    out[(((size_t)b * P_CNT + p) * C_DIM + c0) * 2 + lane] = res;
}

extern "C" void kernel_launch(void* const* d_in, const int* in_sizes, int n_in,
                              void* d_out, int out_size, void* d_ws, size_t ws_size,
                              hipStream_t stream) {
    (void)in_sizes; (void)n_in; (void)out_size; (void)d_ws; (void)ws_size;
    const float* X  = (const float*)d_in[0];
    const float* W1 = (const float*)d_in[1];
    const float* b1 = (const float*)d_in[2];
    const float* W2 = (const float*)d_in[3];
    const float* b2 = (const float*)d_in[4];
    float* out = (float*)d_out;

    const int tiles  = B_DIM * P_CNT * (C_DIM / 16);  // 65,408
    const int blocks = tiles / 8;                      // 8 waves per block
    offset_predictor_kernel<<<blocks, 256, 0, stream>>>(X, W1, b1, W2, b2, out);
}